// GaussianMixture_78537771975377
// MI455X (gfx1250) — compile-verified
//
#include <hip/hip_runtime.h>
#include <cmath>

typedef float v2f __attribute__((ext_vector_type(2)));
typedef float v8f __attribute__((ext_vector_type(8)));

#define TWO_PI_D 6.283185307179586

#define WMMA_F32(A, B, C) \
    __builtin_amdgcn_wmma_f32_16x16x4_f32(false, (A), false, (B), (short)0, (C), false, false)

__device__ __forceinline__ float phi_rsq(float d2, float coef) {
    return __builtin_amdgcn_rsqf(fmaf(fmaxf(d2, 0.0f), coef, 1.0f));
}

// ---------------------------------------------------------------------------
// 1) Squared row norms for X (N rows) and means (K rows)
// ---------------------------------------------------------------------------
__global__ void gmm_norms_kernel(const float* __restrict__ X,
                                 const float* __restrict__ M,
                                 float* __restrict__ normX,
                                 float* __restrict__ normM,
                                 int N, int K, int D) {
    int r = blockIdx.x * blockDim.x + threadIdx.x;
    if (r < N) {
        const float* p = X + (size_t)r * D;
        float s = 0.f;
        for (int d = 0; d < D; ++d) s = fmaf(p[d], p[d], s);
        normX[r] = s;
    } else if (r < N + K) {
        int k = r - N;
        const float* p = M + (size_t)k * D;
        float s = 0.f;
        for (int d = 0; d < D; ++d) s = fmaf(p[d], p[d], s);
        normM[k] = s;
    }
}

// ---------------------------------------------------------------------------
// 2) Prep: softmax(probs), per-k coefficients for C term, and full B term
// ---------------------------------------------------------------------------
__global__ void gmm_prep_kernel(const float* __restrict__ means,
                                const float* __restrict__ variances,
                                const float* __restrict__ logits,
                                float* __restrict__ cc,   // per-k phi coefficient (C term)
                                float* __restrict__ w3,   // per-k weight (C term)
                                float* __restrict__ Bout, // scalar B term
                                int K, int D, int N, float gamma, float four_over_2D3) {
    __shared__ float probs[64];
    __shared__ float red[256];
    int t = threadIdx.x;
    if (t < K) {
        float mx = -1e30f;
        for (int k2 = 0; k2 < K; ++k2) mx = fmaxf(mx, logits[k2]);
        float se = 0.f;
        for (int k2 = 0; k2 < K; ++k2) se += __expf(logits[k2] - mx);
        float p = __expf(logits[t] - mx) / se;
        probs[t] = p;
        float vk = variances[t];
        cc[t] = four_over_2D3 / (2.0f * (vk + 2.0f * gamma));
        w3[t] = 2.0f * p / ((float)N * sqrtf((float)TWO_PI_D * (vk + 2.0f * gamma)));
    }
    __syncthreads();
    float contrib = 0.f;
    if (t < K * K) {
        int i = t / K, j = t % K;
        const float* mi = means + (size_t)i * D;
        const float* mj = means + (size_t)j * D;
        float d2 = 0.f;
        for (int d = 0; d < D; ++d) { float df = mi[d] - mj[d]; d2 = fmaf(df, df, d2); }
        d2 = fmaxf(d2, 0.f);
        float vs = variances[i] + variances[j];
        float b2 = __builtin_amdgcn_rsqf(1.0f + four_over_2D3 * d2 / (2.0f * vs + 4.0f * gamma));
        float b3 = probs[i] * probs[j] *
                   __builtin_amdgcn_rsqf((float)TWO_PI_D * (vs + 2.0f * gamma));
        contrib = b2 * b3;
    }
    red[t] = contrib;
    __syncthreads();
    for (int s = 128; s > 0; s >>= 1) {
        if (t < s) red[t] += red[t + s];
        __syncthreads();
    }
    if (t == 0) Bout[0] = red[0];
}

// ---------------------------------------------------------------------------
// 3) A term: fused syrk(X) + phi + reduce via V_WMMA_F32_16X16X4_F32.
//    One wave per 32x32 block (2x2 register-blocked 16x16 WMMA tiles).
//    Chunked prefetch: 32 b64 loads, one wait, 32 back-to-back WMMAs.
//    Block-level triangular symmetry (diag blocks w=1, off-diag w=2).
//    Requires N % 32 == 0 and D % 32 == 0 (16384, 64 per reference).
// ---------------------------------------------------------------------------
__global__ void gmm_gram_phi_kernel(const float* __restrict__ X,
                                    const float* __restrict__ normX,
                                    float* __restrict__ partials,
                                    int NTB, int D, float c1) {
    const int lane = threadIdx.x & 31;
    const int wave = threadIdx.x >> 5;
    const int wpb  = blockDim.x >> 5;
    const int gw   = blockIdx.x * wpb + wave;
    const int nw   = gridDim.x * wpb;
    const int m    = lane & 15;
    const int hh   = lane >> 4;      // which K-pair this lane holds
    const long total = (long)NTB * (long)NTB;

    float sum = 0.f;
    for (long t = gw; t < total; t += nw) {
        int ib = (int)(t / NTB);
        int jb = (int)(t - (long)ib * NTB);
        if (jb < ib) continue;       // wave-uniform: EXEC stays all-ones
        const float* rA0 = X + (size_t)(ib * 32 + m) * D + 2 * hh;
        const float* rA1 = rA0 + (size_t)16 * D;
        const float* rB0 = X + (size_t)(jb * 32 + m) * D + 2 * hh;
        const float* rB1 = rB0 + (size_t)16 * D;

        v8f acc00 = {}, acc01 = {}, acc10 = {}, acc11 = {};
        for (int kk = 0; kk < D; kk += 32) {
            v2f a0[8], a1[8], b0[8], b1[8];
#pragma unroll
            for (int q = 0; q < 8; ++q) {
                a0[q] = *(const v2f*)(rA0 + kk + 4 * q);
                a1[q] = *(const v2f*)(rA1 + kk + 4 * q);
                b0[q] = *(const v2f*)(rB0 + kk + 4 * q);
                b1[q] = *(const v2f*)(rB1 + kk + 4 * q);
            }
#pragma unroll
            for (int q = 0; q < 8; ++q) {
                acc00 = WMMA_F32(a0[q], b0[q], acc00);
                acc01 = WMMA_F32(a0[q], b1[q], acc01);
                acc10 = WMMA_F32(a1[q], b0[q], acc10);
                acc11 = WMMA_F32(a1[q], b1[q], acc11);
            }
        }

        float w   = (ib == jb) ? 1.0f : 2.0f;
        float nB0 = normX[jb * 32 + m];        // col index == m (tiles j0)
        float nB1 = normX[jb * 32 + 16 + m];   // tiles j1
        float ts  = 0.f;
#pragma unroll
        for (int v = 0; v < 8; ++v) {
            int   r0  = ib * 32 + v + 8 * hh;
            float nA0 = normX[r0];
            float nA1 = normX[r0 + 16];
            ts += phi_rsq(nA0 + nB0 - 2.0f * acc00[v], c1);
            ts += phi_rsq(nA0 + nB1 - 2.0f * acc01[v], c1);
            ts += phi_rsq(nA1 + nB0 - 2.0f * acc10[v], c1);
            ts += phi_rsq(nA1 + nB1 - 2.0f * acc11[v], c1);
        }
        sum += w * ts;
    }

    __shared__ float red[256];
    red[threadIdx.x] = sum;
    __syncthreads();
    for (int s = blockDim.x / 2; s > 0; s >>= 1) {
        if ((int)threadIdx.x < s) red[threadIdx.x] += red[threadIdx.x + s];
        __syncthreads();
    }
    if (threadIdx.x == 0) partials[blockIdx.x] = red[0];
}

// ---------------------------------------------------------------------------
// 4) C term: fused X @ means^T + phi + weighted reduce, chunked prefetch
// ---------------------------------------------------------------------------
__global__ void gmm_cross_phi_kernel(const float* __restrict__ X,
                                     const float* __restrict__ means,
                                     const float* __restrict__ normX,
                                     const float* __restrict__ normM,
                                     const float* __restrict__ cc,
                                     const float* __restrict__ w3,
                                     float* __restrict__ partials,
                                     int NT, int KT, int D) {
    const int lane = threadIdx.x & 31;
    const int wave = threadIdx.x >> 5;
    const int wpb  = blockDim.x >> 5;
    const int gw   = blockIdx.x * wpb + wave;
    const int nw   = gridDim.x * wpb;
    const int m    = lane & 15;
    const int hh   = lane >> 4;
    const int total = NT * KT;

    float sum = 0.f;
    for (int t = gw; t < total; t += nw) {
        int i = t / KT;
        int j = t - i * KT;
        const float* rowA = X + (size_t)(i * 16 + m) * D + 2 * hh;
        const float* rowB = means + (size_t)(j * 16 + m) * D + 2 * hh;
        v8f acc = {};
        for (int kk = 0; kk < D; kk += 32) {
            v2f a[8], b[8];
#pragma unroll
            for (int q = 0; q < 8; ++q) {
                a[q] = *(const v2f*)(rowA + kk + 4 * q);
                b[q] = *(const v2f*)(rowB + kk + 4 * q);
            }
#pragma unroll
            for (int q = 0; q < 8; ++q) acc = WMMA_F32(a[q], b[q], acc);
        }
        int   k    = j * 16 + m;               // column index
        float nM   = normM[k];
        float coef = cc[k];
        float wk   = w3[k];
#pragma unroll
        for (int v = 0; v < 8; ++v) {
            int row = v + 8 * hh;
            sum += wk * phi_rsq(normX[i * 16 + row] + nM - 2.0f * acc[v], coef);
        }
    }

    __shared__ float red[256];
    red[threadIdx.x] = sum;
    __syncthreads();
    for (int s = blockDim.x / 2; s > 0; s >>= 1) {
        if ((int)threadIdx.x < s) red[threadIdx.x] += red[threadIdx.x + s];
        __syncthreads();
    }
    if (threadIdx.x == 0) partials[blockIdx.x] = red[0];
}

// ---------------------------------------------------------------------------
// 5) Finalize: deterministic double-precision reduce + log(A + B - C)
// ---------------------------------------------------------------------------
__global__ void gmm_finalize_kernel(const float* __restrict__ Apart, int nA,
                                    const float* __restrict__ Cpart, int nC,
                                    const float* __restrict__ Bsum,
                                    float* __restrict__ out, double scaleA) {
    __shared__ double red[256];
    int t = threadIdx.x;
    double sA = 0.0;
    for (int i = t; i < nA; i += blockDim.x) sA += (double)Apart[i];
    red[t] = sA;
    __syncthreads();
    for (int s = 128; s > 0; s >>= 1) { if (t < s) red[t] += red[t + s]; __syncthreads(); }
    double A1 = red[0];
    __syncthreads();
    double sC = 0.0;
    for (int i = t; i < nC; i += blockDim.x) sC += (double)Cpart[i];
    red[t] = sC;
    __syncthreads();
    for (int s = 128; s > 0; s >>= 1) { if (t < s) red[t] += red[t + s]; __syncthreads(); }
    if (t == 0) {
        double A = A1 * scaleA;
        double B = (double)Bsum[0];
        double C = red[0];
        out[0] = (float)log(A + B - C);
    }
}

// ---------------------------------------------------------------------------
extern "C" void kernel_launch(void* const* d_in, const int* in_sizes, int n_in,
                              void* d_out, int out_size, void* d_ws, size_t ws_size,
                              hipStream_t stream) {
    const float* X         = (const float*)d_in[0];
    const float* means     = (const float*)d_in[1];
    const float* variances = (const float*)d_in[2];
    const float* logits    = (const float*)d_in[3];
    float* out = (float*)d_out;

    const int K = in_sizes[2];          // 16
    const int D = in_sizes[1] / K;      // 64
    const int N = in_sizes[0] / D;      // 16384
    const int NTB = N / 32;             // 32-row blocks for the syrk kernel
    const int NT = N / 16, KT = K / 16;

    const double gamma  = pow(4.0 / (3.0 * (double)N / (double)K), 0.4);
    const double twoD3  = 2.0 * (double)D - 3.0;                // 125
    const float  c1     = (float)(1.0 / (twoD3 * gamma));       // A: rsq(1 + d2*c1)
    const float  f2d3   = (float)(4.0 / twoD3);                 // 4/(2D-3)
    const double scaleA = 1.0 / ((double)N * (double)N * sqrt(TWO_PI_D * 2.0 * gamma));

    const int NA = 4096;   // A partial blocks
    const int NC = 64;     // C partial blocks
    float* ws    = (float*)d_ws;
    float* Apart = ws;                    // [NA]
    float* Cpart = ws + NA;               // [NC]
    float* Bsum  = ws + NA + NC;          // [1]
    float* cc    = ws + NA + NC + 64;     // [K] (padded start)
    float* w3    = cc + K;                // [K]
    float* normX = w3 + K;                // [N]
    float* normM = normX + N;             // [K]

    gmm_norms_kernel<<<(N + K + 255) / 256, 256, 0, stream>>>(X, means, normX, normM, N, K, D);
    gmm_prep_kernel<<<1, 256, 0, stream>>>(means, variances, logits, cc, w3, Bsum,
                                           K, D, N, (float)gamma, f2d3);
    gmm_gram_phi_kernel<<<NA, 256, 0, stream>>>(X, normX, Apart, NTB, D, c1);
    gmm_cross_phi_kernel<<<NC, 256, 0, stream>>>(X, means, normX, normM, cc, w3, Cpart,
                                                 NT, KT, D);
    gmm_finalize_kernel<<<1, 256, 0, stream>>>(Apart, NA, Cpart, NC, Bsum, out, scaleA);
}